// EMRGNN_68470368633607
// MI455X (gfx1250) — compile-verified
//
#include <hip/hip_runtime.h>
#include <math.h>

typedef __attribute__((ext_vector_type(2))) float v2f;
typedef __attribute__((ext_vector_type(8))) float v8f;

#define NNODES   30000
#define RREL     5
#define NEDGE    600000
#define HDIM     128
#define L1C      20.0f
#define RATIO    3.0f      // 2*L2/L1
#define THRESH_V 0.3f
#define THRESH_C 1e-3f
#define T_INNER  20
#define K_OUTER  8
#define EPSLN    1e-5f

// ---------------------------------------------------------------------------
// Dense fp32 GEMM with V_WMMA_F32_16X16X4_F32.
// One block = 16 rows x 128 cols of Y. 8 waves, wave w owns cols [16w,16w+16).
// epi: 0 = bias only; 1 = bias + layernorm(ddof=0) + relu; 2 = bias +
// row-standardize (ddof=1), sd==0 -> 0 (nan_to_num).
// M is a multiple of 16 and K a multiple of 4 for every call site.
// ---------------------------------------------------------------------------
__global__ __launch_bounds__(256)
void k_gemm128(const float* __restrict__ X, const float* __restrict__ W,
               const float* __restrict__ bias, float* __restrict__ Y,
               int K, int epi)
{
    __shared__ float tile[16][132];
    __shared__ float smu[16], sinv[16];
    const int lane = threadIdx.x & 31;
    const int wv   = threadIdx.x >> 5;
    const int half = lane >> 4;          // 0: lanes 0-15, 1: lanes 16-31
    const int l16  = lane & 15;
    const int row0 = blockIdx.x << 4;
    const int bcol = (wv << 4) + l16;    // output column 0..127
    const float* xr = X + (size_t)(row0 + l16) * K;

    v8f acc = {};
    for (int k0 = 0; k0 < K; k0 += 4) {
        const int ka = k0 + half * 2;    // lanes 0-15: K=k0,k0+1; 16-31: k0+2,k0+3
        v2f a, b;
        a.x = xr[ka];
        a.y = xr[ka + 1];
        b.x = W[(size_t)ka * HDIM + bcol];
        b.y = W[(size_t)(ka + 1) * HDIM + bcol];
        acc = __builtin_amdgcn_wmma_f32_16x16x4_f32(false, a, false, b,
                                                    (short)0, acc, false, false);
    }
    const float bv = bias[bcol];
#pragma unroll
    for (int i = 0; i < 8; ++i)
        tile[i + half * 8][bcol] = acc[i] + bv;   // M = i + 8*half, N = bcol
    __syncthreads();

    if (epi) {
        // wave wv reduces rows 2wv and 2wv+1 over 128 cols
        for (int r = wv * 2; r < wv * 2 + 2; ++r) {
            float s1 = 0.f, s2 = 0.f;
#pragma unroll
            for (int j = 0; j < 4; ++j) {
                float v = tile[r][lane + 32 * j];
                s1 += v; s2 += v * v;
            }
#pragma unroll
            for (int off = 16; off; off >>= 1) {
                s1 += __shfl_xor(s1, off);
                s2 += __shfl_xor(s2, off);
            }
            if (lane == 0) {
                float mu = s1 * (1.0f / 128.0f);
                float inv;
                if (epi == 1) {                              // LN, ddof=0
                    float var = fmaxf(s2 * (1.0f / 128.0f) - mu * mu, 0.0f);
                    inv = 1.0f / sqrtf(var + EPSLN);
                } else {                                     // std, ddof=1
                    float var1 = fmaxf(s2 - 128.0f * mu * mu, 0.0f) * (1.0f / 127.0f);
                    float sd = sqrtf(var1);
                    inv = (sd > 0.0f) ? (1.0f / sd) : 0.0f;  // nan_to_num -> 0
                }
                smu[r] = mu; sinv[r] = inv;
            }
        }
        __syncthreads();
    }
    // coalesced store, 8 elems/thread
    for (int j = 0; j < 8; ++j) {
        int idx = (int)threadIdx.x + 256 * j;
        int r = idx >> 7, c = idx & 127;
        float v = tile[r][c];
        if (epi) {
            v = (v - smu[r]) * sinv[r];
            if (epi == 1) v = fmaxf(v, 0.0f);
        }
        Y[(size_t)(row0 + r) * HDIM + c] = v;
    }
}

// logits = x @ Wout + bout   (K=128, 8 output cols; WMMA with cols 8..15 = 0)
__global__ __launch_bounds__(32)
void k_gemm_logits(const float* __restrict__ X, const float* __restrict__ W,
                   const float* __restrict__ b, float* __restrict__ Out)
{
    const int lane = threadIdx.x;
    const int half = lane >> 4, l16 = lane & 15;
    const int row0 = blockIdx.x << 4;
    const float* xr = X + (size_t)(row0 + l16) * HDIM;
    v8f acc = {};
    for (int k0 = 0; k0 < HDIM; k0 += 4) {
        const int ka = k0 + half * 2;
        v2f a, bb;
        a.x = xr[ka]; a.y = xr[ka + 1];
        bb.x = (l16 < 8) ? W[(size_t)ka * 8 + l16] : 0.0f;
        bb.y = (l16 < 8) ? W[(size_t)(ka + 1) * 8 + l16] : 0.0f;
        acc = __builtin_amdgcn_wmma_f32_16x16x4_f32(false, a, false, bb,
                                                    (short)0, acc, false, false);
    }
    if (l16 < 8) {
        const float bv = b[l16];
#pragma unroll
        for (int i = 0; i < 8; ++i)
            Out[(size_t)(row0 + i + half * 8) * 8 + l16] = acc[i] + bv;
    }
}

// --------------------------- graph helper kernels ---------------------------
__global__ void k_zero(float* __restrict__ p, int n)
{
    int i = blockIdx.x * blockDim.x + threadIdx.x;
    int stride = gridDim.x * blockDim.x;
    for (; i < n; i += stride) p[i] = 0.0f;
}

__global__ void k_copy(float* __restrict__ o, const float* __restrict__ in, int n)
{
    int i = blockIdx.x * blockDim.x + threadIdx.x;
    int stride = gridDim.x * blockDim.x;
    for (; i < n; i += stride) o[i] = in[i];
}

__global__ void k_deg(const int* __restrict__ src, const int* __restrict__ rel,
                      float* __restrict__ deg, int e_)
{
    int e = blockIdx.x * blockDim.x + threadIdx.x;
    if (e < e_) atomicAdd(&deg[rel[e] * NNODES + src[e]], 1.0f);
}

__global__ void k_dinv(const float* __restrict__ deg, float* __restrict__ dinv, int n)
{
    int i = blockIdx.x * blockDim.x + threadIdx.x;
    if (i < n) {
        float d = deg[i];
        dinv[i] = (d > 0.0f) ? (1.0f / sqrtf(d)) : 0.0f;
    }
}

__global__ void k_edge_pre(const int* __restrict__ src, const int* __restrict__ dst,
                           const int* __restrict__ rel, const float* __restrict__ deg,
                           const float* __restrict__ dinv, float* __restrict__ vals,
                           float* __restrict__ edinv, int e_)
{
    int e = blockIdx.x * blockDim.x + threadIdx.x;
    if (e >= e_) return;
    int r  = rel[e];
    int ks = r * NNODES + src[e];
    float d = deg[ks];
    vals[e]  = (d > 0.0f) ? (1.0f / d) : 0.0f;
    edinv[e] = dinv[ks] * dinv[r * NNODES + dst[e]];
}

__global__ void k_rowsq(const float* __restrict__ x, float* __restrict__ s, int n)
{
    int i = blockIdx.x * blockDim.x + threadIdx.x;
    if (i < n) {
        const float4* p = (const float4*)(x + (size_t)i * HDIM);
        float a = 0.0f;
#pragma unroll
        for (int j = 0; j < 32; ++j) {
            float4 v = p[j];
            a += v.x * v.x + v.y * v.y + v.z * v.z + v.w * v.w;
        }
        s[i] = a;
    }
}

__global__ __launch_bounds__(256)
void k_node_term(const float* __restrict__ deg, const float* __restrict__ s,
                 float* __restrict__ nterm, int n)
{
    __shared__ float red[256];
    int r = blockIdx.y;
    int i = blockIdx.x * 256 + threadIdx.x;
    float v = 0.0f;
    if (i < n && deg[r * NNODES + i] > 0.0f) v = s[i];
    red[threadIdx.x] = v;
    __syncthreads();
    for (int off = 128; off; off >>= 1) {
        if ((int)threadIdx.x < off) red[threadIdx.x] += red[threadIdx.x + off];
        __syncthreads();
    }
    if (threadIdx.x == 0) atomicAdd(&nterm[r], red[0]);
}

// one wave per edge, 8 edges per wave -> 64 edges per block
__global__ __launch_bounds__(256)
void k_edge_term(const float* __restrict__ x, const int* __restrict__ src,
                 const int* __restrict__ dst, const int* __restrict__ rel,
                 const float* __restrict__ edinv, float* __restrict__ eterm, int e_)
{
    __shared__ float sh[8];
    if (threadIdx.x < 8) sh[threadIdx.x] = 0.0f;
    __syncthreads();
    const int lane = threadIdx.x & 31;
    const int wv   = threadIdx.x >> 5;
    const int base = blockIdx.x * 64 + wv * 8;
    for (int i = 0; i < 8; ++i) {
        int e = base + i;
        if (e < e_) {
            const float* xs = x + (size_t)src[e] * HDIM;
            const float* xd = x + (size_t)dst[e] * HDIM;
            float acc = 0.0f;
#pragma unroll
            for (int j = 0; j < 4; ++j) {
                int c = lane + 32 * j;
                acc += xs[c] * xd[c];
            }
#pragma unroll
            for (int off = 16; off; off >>= 1) acc += __shfl_xor(acc, off);
            if (lane == 0) atomicAdd(&sh[rel[e]], edinv[e] * acc);
        }
    }
    __syncthreads();
    if (threadIdx.x < RREL) atomicAdd(&eterm[threadIdx.x], sh[threadIdx.x]);
}

// control block cb: [0..4]=u, [5]=c_l1tr, [6]=act, [7]=alpha, [8]=beta
__global__ void k_init_control(const float* __restrict__ nt, const float* __restrict__ et,
                               float* __restrict__ cb)
{
    if (threadIdx.x != 0 || blockIdx.x != 0) return;
    float l1 = 0.0f;
    for (int r = 0; r < RREL; ++r)
        l1 += fabsf((nt[r] - et[r]) * (1.0f / (float)NNODES));
    cb[5] = l1;
    for (int r = 0; r < RREL; ++r) cb[r] = 1.0f / (float)RREL;
    cb[6] = 1.0f; cb[7] = 1.0f; cb[8] = 0.0f;
}

__global__ void k_control(const float* __restrict__ nt, const float* __restrict__ et,
                          float* __restrict__ cb)
{
    if (threadIdx.x != 0 || blockIdx.x != 0) return;
    float w[RREL], u[RREL];
    float l1 = 0.0f;
    for (int r = 0; r < RREL; ++r) {
        w[r] = (nt[r] - et[r]) * (1.0f / (float)NNODES);
        l1 += fabsf(w[r]);
        u[r] = cb[r];
    }
    bool act = cb[6] > 0.5f;
    if (act) {
        float fi = l1 + RATIO;
        bool ia = true;
        for (int t = 1; t <= T_INNER; ++t) {
            float Tt = sqrtf(2.0f * logf((float)RREL) / ((float)t * fi * fi));
            float ta[RREL], ssum = 0.0f;
            for (int r = 0; r < RREL; ++r) {
                ta[r] = u[r] * expf(-Tt * (RATIO * u[r] + w[r]));
                ssum += ta[r];
            }
            float d2 = 0.0f;
            for (int r = 0; r < RREL; ++r) {
                float un = ta[r] / ssum;
                float dd = u[r] - un;
                d2 += dd * dd;
                ta[r] = un;
            }
            float diff = sqrtf(d2);
            if (ia) for (int r = 0; r < RREL; ++r) u[r] = ta[r];
            ia = ia && (diff >= THRESH_C);
        }
        for (int r = 0; r < RREL; ++r) cb[r] = u[r];
    }
    bool still = act && (l1 / cb[5] >= THRESH_V);
    cb[6] = still ? 1.0f : 0.0f;
    cb[7] = still ? (1.0f / (1.0f + L1C)) : 1.0f;
    cb[8] = still ? (L1C / (1.0f + L1C)) : 0.0f;
}

// one wave per edge; scatter msg[src] += u[rel]*vals * x[dst]
__global__ __launch_bounds__(256)
void k_spmm(const float* __restrict__ x, const int* __restrict__ src,
            const int* __restrict__ dst, const int* __restrict__ rel,
            const float* __restrict__ vals, const float* __restrict__ cb,
            float* __restrict__ msg, int e_)
{
    int gw   = (int)((blockIdx.x * 256 + threadIdx.x) >> 5);
    int lane = threadIdx.x & 31;
    if (gw >= e_) return;
    int s = src[gw], d = dst[gw];
    float ew = cb[rel[gw]] * vals[gw];
    if (ew != 0.0f) {
        const float* xd = x + (size_t)d * HDIM;
        float* ms = msg + (size_t)s * HDIM;
#pragma unroll
        for (int j = 0; j < 4; ++j) {
            int c = lane + 32 * j;
            atomicAdd(&ms[c], ew * xd[c]);
        }
    }
}

__global__ void k_axpy(float* __restrict__ x, const float* __restrict__ msg,
                       const float* __restrict__ cb, int n)
{
    int i = blockIdx.x * blockDim.x + threadIdx.x;
    int stride = gridDim.x * blockDim.x;
    float alpha = cb[7], beta = cb[8];       // (1,0) when !still -> x unchanged
    for (; i < n; i += stride) x[i] = alpha * x[i] + beta * msg[i];
}

// ---------------------------------------------------------------------------
extern "C" void kernel_launch(void* const* d_in, const int* in_sizes, int n_in,
                              void* d_out, int out_size, void* d_ws, size_t ws_size,
                              hipStream_t stream)
{
    const float* feat0 = (const float*)d_in[0];
    const float* feat1 = (const float*)d_in[1];
    const float* W0  = (const float*)d_in[2];  const float* b0  = (const float*)d_in[3];
    const float* W1  = (const float*)d_in[4];  const float* b1  = (const float*)d_in[5];
    const float* Wm1 = (const float*)d_in[6];  const float* bm1 = (const float*)d_in[7];
    const float* Wm2 = (const float*)d_in[8];  const float* bm2 = (const float*)d_in[9];
    const float* Wout= (const float*)d_in[10]; const float* bout= (const float*)d_in[11];
    const int* src = (const int*)d_in[12];
    const int* dst = (const int*)d_in[13];
    const int* rel = (const int*)d_in[14];
    (void)in_sizes; (void)n_in; (void)out_size; (void)ws_size;

    const int NB = NNODES * HDIM;            // 3,840,000
    float* ws   = (float*)d_ws;              // ~52.2 MB used
    float* H    = ws;                        // (N,128) hidden; reused as MSG
    float* Z    = ws + (size_t)NB;           // (N,128)
    float* X    = ws + (size_t)2 * NB;       // (N,128) current x
    float* S    = ws + (size_t)3 * NB;       // (N,) row sumsq
    float* DEG  = S + NNODES;                // (R*N,)
    float* DINV = DEG + RREL * NNODES;       // (R*N,)
    float* VALS = DINV + RREL * NNODES;      // (E,)
    float* EDI  = VALS + NEDGE;              // (E,)
    float* NT   = EDI + NEDGE;               // 8
    float* ET   = NT + 8;                    // 8 (contiguous with NT)
    float* CB   = ET + 8;                    // 16: u[5], c_l1tr, act, alpha, beta
    float* OUT  = (float*)d_out;             // [logits 240000][x 3840000][u 5]

    // ---- dense transfer chain (WMMA fp32) ----
    k_gemm128<<<18000 / 16, 256, 0, stream>>>(feat0, W0, b0, H, 256, 0);
    k_gemm128<<<12000 / 16, 256, 0, stream>>>(feat1, W1, b1, H + (size_t)18000 * HDIM, 128, 0);
    k_gemm128<<<NNODES / 16, 256, 0, stream>>>(H, Wm1, bm1, Z, 128, 1);   // LN+ReLU
    k_gemm128<<<NNODES / 16, 256, 0, stream>>>(Z, Wm2, bm2, X, 128, 2);   // standardize

    // ---- graph preprocessing ----
    k_zero<<<(RREL * NNODES + 255) / 256, 256, 0, stream>>>(DEG, RREL * NNODES);
    k_deg<<<(NEDGE + 255) / 256, 256, 0, stream>>>(src, rel, DEG, NEDGE);
    k_dinv<<<(RREL * NNODES + 255) / 256, 256, 0, stream>>>(DEG, DINV, RREL * NNODES);
    k_edge_pre<<<(NEDGE + 255) / 256, 256, 0, stream>>>(src, dst, rel, DEG, DINV, VALS, EDI, NEDGE);

    // ---- c_l1tr from tv(out0) ----
    k_rowsq<<<(NNODES + 255) / 256, 256, 0, stream>>>(X, S, NNODES);
    k_zero<<<1, 32, 0, stream>>>(NT, 16);
    k_node_term<<<dim3((NNODES + 255) / 256, RREL), 256, 0, stream>>>(DEG, S, NT, NNODES);
    k_edge_term<<<(NEDGE + 63) / 64, 256, 0, stream>>>(X, src, dst, rel, EDI, ET, NEDGE);
    k_init_control<<<1, 32, 0, stream>>>(NT, ET, CB);

    // ---- outer loop ----
    for (int k = 0; k < K_OUTER; ++k) {
        k_rowsq<<<(NNODES + 255) / 256, 256, 0, stream>>>(X, S, NNODES);
        k_zero<<<1, 32, 0, stream>>>(NT, 16);
        k_node_term<<<dim3((NNODES + 255) / 256, RREL), 256, 0, stream>>>(DEG, S, NT, NNODES);
        k_edge_term<<<(NEDGE + 63) / 64, 256, 0, stream>>>(X, src, dst, rel, EDI, ET, NEDGE);
        k_control<<<1, 32, 0, stream>>>(NT, ET, CB);
        k_zero<<<1875, 256, 0, stream>>>(H, NB);                 // MSG = 0
        k_spmm<<<(NEDGE + 7) / 8, 256, 0, stream>>>(X, src, dst, rel, VALS, CB, H, NEDGE);
        k_axpy<<<1875, 256, 0, stream>>>(X, H, CB, NB);
    }

    // ---- outputs ----
    k_gemm_logits<<<NNODES / 16, 32, 0, stream>>>(X, Wout, bout, OUT);
    k_copy<<<1875, 256, 0, stream>>>(OUT + 240000, X, NB);
    k_copy<<<1, 32, 0, stream>>>(OUT + 240000 + NB, CB, RREL);
}